// IntraConv_multi_75591424410316
// MI455X (gfx1250) — compile-verified
//
#include <hip/hip_runtime.h>
#include <hip/hip_bf16.h>

// GraphSAGE mean-agg layer for MI455X (gfx1250, wave32).
//   out = feat @ W_self^T + b_self + mean_by_dst(feat[src]) @ W_neigh^T + bias
//
// Cost model: scatter (~410MB L2-resident gathers + ~102M f32 L2 atomics)
// dominates. GEMM fused into one K=256 f32 WMMA chain; W matrices staged in
// LDS once per block (8 M-tiles/block) so B traffic is ~50MB instead of
// ~400MB; LDS rows padded to 132 floats to kill 16-way bank conflicts.

#define NN 50000     // nodes
#define DD 128       // feature dim (in == out)
#define WST 132      // padded LDS row stride in floats (132 mod 64 = 4)
#define TPB_TILES 8  // M-tiles (16 rows each) per block -> 391 blocks

typedef __attribute__((ext_vector_type(2))) float v2f;
typedef __attribute__((ext_vector_type(8))) float v8f;

// ---------------- kernel 1: zero the accumulation workspace ----------------
__global__ void zero_ws_kernel(float4* __restrict__ p, int n4) {
  int i = blockIdx.x * blockDim.x + threadIdx.x;
  const int stride = gridDim.x * blockDim.x;
  const float4 z = {0.f, 0.f, 0.f, 0.f};
  for (; i < n4; i += stride) p[i] = z;
}

// ---------------- kernel 2: edge scatter (wave per edge) -------------------
// Each of 32 lanes handles 4 consecutive floats of the 128-wide row:
// one b128 gather from feat[src] + 4 global f32 atomic adds into nsum[dst].
__global__ void edge_scatter_kernel(const float* __restrict__ feat,
                                    const int*   __restrict__ src,
                                    const int*   __restrict__ dst,
                                    float* __restrict__ nsum,
                                    float* __restrict__ deg,
                                    int E) {
  const int gid  = blockIdx.x * blockDim.x + threadIdx.x;
  const int e    = gid >> 5;
  const int lane = threadIdx.x & 31;
  if (e >= E) return;
  const int s = src[e];          // uniform across the wave -> one L2 request
  const int d = dst[e];
  const float4 v = ((const float4*)(feat + (size_t)s * DD))[lane];
  float* o = nsum + (size_t)d * DD + lane * 4;
  atomicAdd(o + 0, v.x);
  atomicAdd(o + 1, v.y);
  atomicAdd(o + 2, v.z);
  atomicAdd(o + 3, v.w);
  if (lane == 0) atomicAdd(deg + d, 1.0f);
}

// ---------------- kernel 3: fused WMMA GEMM --------------------------------
// Block = 256 threads = 8 waves; wave w owns N-tile w (16 output cols).
// Block stages W_self+W_neigh into padded LDS once, then loops over
// TPB_TILES M-tiles, staging each 16-row A slab (feat + scaled neigh) into
// padded LDS and running the K=2x128 f32 WMMA chain entirely out of LDS.
__global__ __launch_bounds__(256)
void sage_gemm_kernel(const float* __restrict__ feat,
                      const float* __restrict__ nsum,
                      const float* __restrict__ deg,
                      const float* __restrict__ Wself,
                      const float* __restrict__ bself,
                      const float* __restrict__ Wneigh,
                      const float* __restrict__ bias,
                      float* __restrict__ out) {
  __shared__ float sWs[DD * WST];  // Wself rows: [col][k], padded
  __shared__ float sWn[DD * WST];  // Wneigh rows
  __shared__ float sF[16 * WST];   // feat slab
  __shared__ float sN[16 * WST];   // neigh-mean slab (scaled while staging)
  const int tid = threadIdx.x;

  // ---- stage both weight matrices into LDS (one pass per block) ----
  {
    const float4* gWs = (const float4*)Wself;
    const float4* gWn = (const float4*)Wneigh;
    for (int i = tid; i < DD * (DD / 4); i += 256) {
      const int row = i >> 5;         // W row = output col
      const int c4  = i & 31;
      *(float4*)&sWs[row * WST + c4 * 4] = gWs[i];
      *(float4*)&sWn[row * WST + c4 * 4] = gWn[i];
    }
  }

  const int wave = tid >> 5;       // N-tile id, 0..7
  const int lane = tid & 31;
  const int r    = lane & 15;
  const int hi   = lane >> 4;
  const int col  = wave * 16 + r;
  const float* __restrict__ Bs = sWs + (size_t)col * WST;  // B[k][col]=W[col][k]
  const float* __restrict__ Bn = sWn + (size_t)col * WST;
  const float* __restrict__ As = sF + r * WST;
  const float* __restrict__ An = sN + r * WST;
  const float bsum = bself[col] + bias[col];

  const int tile0   = blockIdx.x * TPB_TILES;
  const int tileEnd = (tile0 + TPB_TILES < NN / 16) ? tile0 + TPB_TILES
                                                    : NN / 16;
  for (int mt = tile0; mt < tileEnd; ++mt) {
    const int m0 = mt * 16;
    __syncthreads();   // waves done reading previous slab (and W on iter 0)

    // ---- stage 16-row A slab: feat + nsum/max(deg,1), padded stride ----
    {
      const float4* gF = (const float4*)(feat + (size_t)m0 * DD);
      const float4* gN = (const float4*)(nsum + (size_t)m0 * DD);
      for (int i = tid; i < 16 * (DD / 4); i += 256) {
        const int row = i >> 5;
        const int c4  = i & 31;
        *(float4*)&sF[row * WST + c4 * 4] = gF[i];
        const float dg = deg[m0 + row];
        const float sc = 1.0f / fmaxf(dg, 1.0f);   // deg==0 => sum==0 => 0
        float4 v = gN[i];
        v.x *= sc; v.y *= sc; v.z *= sc; v.w *= sc;
        *(float4*)&sN[row * WST + c4 * 4] = v;
      }
    }
    __syncthreads();

    // ---- prefetch next A slab into cache while we compute ----
    if (mt + 1 < tileEnd) {
      if (tid < 64) {          // 64 lanes x 32 floats = 2048 floats = slab
        __builtin_prefetch(feat + (size_t)(m0 + 16) * DD + tid * 32, 0, 0);
      } else if (tid < 128) {
        __builtin_prefetch(nsum + (size_t)(m0 + 16) * DD + (tid - 64) * 32,
                           0, 0);
      }
    }

    // ---- K = 2x128 fused f32 WMMA chain, all operands from LDS ----
    // A 16x4 layout: lanes 0-15 K={k,k+1}, lanes 16-31 K={k+2,k+3}, M=r.
    // B 4x16 layout mirrors: VGPR j holds K=k+hi*2+j, N=r.
    v8f acc = {};
    for (int k = 0; k < DD; k += 4) {
      const int ka = k + hi * 2;
      v2f a, b;
      a.x = As[ka]; a.y = As[ka + 1];
      b.x = Bs[ka]; b.y = Bs[ka + 1];
      acc = __builtin_amdgcn_wmma_f32_16x16x4_f32(false, a, false, b,
                                                  (short)0, acc, false, false);
      a.x = An[ka]; a.y = An[ka + 1];
      b.x = Bn[ka]; b.y = Bn[ka + 1];
      acc = __builtin_amdgcn_wmma_f32_16x16x4_f32(false, a, false, b,
                                                  (short)0, acc, false, false);
    }

    // C/D layout: VGPR rr -> M = rr + hi*8, N = col
    #pragma unroll
    for (int rr = 0; rr < 8; ++rr) {
      const int row = m0 + hi * 8 + rr;
      out[(size_t)row * DD + col] = acc[rr] + bsum;
    }
  }
}

extern "C" void kernel_launch(void* const* d_in, const int* in_sizes, int n_in,
                              void* d_out, int out_size, void* d_ws, size_t ws_size,
                              hipStream_t stream) {
  const float* feat   = (const float*)d_in[0];
  const int*   src    = (const int*)d_in[1];
  const int*   dst    = (const int*)d_in[2];
  const float* Wself  = (const float*)d_in[3];
  const float* bself  = (const float*)d_in[4];
  const float* Wneigh = (const float*)d_in[5];
  const float* bias   = (const float*)d_in[6];
  float* out = (float*)d_out;
  const int E = in_sizes[1];

  // workspace: [NN*DD] f32 neighbor sums, then [NN] f32 degrees (~25.8 MB)
  float* nsum = (float*)d_ws;
  float* deg  = nsum + (size_t)NN * DD;

  const int n4 = (NN * DD + NN) / 4;               // 1,612,500 float4s
  zero_ws_kernel<<<2048, 256, 0, stream>>>((float4*)d_ws, n4);

  edge_scatter_kernel<<<(E + 7) / 8, 256, 0, stream>>>(feat, src, dst,
                                                       nsum, deg, E);

  const int ntiles = NN / 16;                      // 3125
  const int gblocks = (ntiles + TPB_TILES - 1) / TPB_TILES;  // 391
  sage_gemm_kernel<<<gblocks, 256, 0, stream>>>(feat, nsum, deg, Wself, bself,
                                                Wneigh, bias, out);
}